// HGAN_2095944040715
// MI455X (gfx1250) — compile-verified
//
#include <hip/hip_runtime.h>
#include <math.h>
#include <stdint.h>

typedef float v2f __attribute__((ext_vector_type(2)));
typedef float v8f __attribute__((ext_vector_type(8)));

#define EDIM 768
#define HDIM 512
#define KH   8
#define NBAT 32
#define NAA  64
#define NCC  64

static __device__ __forceinline__ v8f wmma_f32(v2f a, v2f b, v8f c) {
    // V_WMMA_F32_16X16X4_F32 : D = A(16x4) * B(4x16) + C(16x16), all f32
    return __builtin_amdgcn_wmma_f32_16x16x4_f32(false, a, false, b, (short)0, c, false, false);
}

// Async global->LDS copy of 32 bytes per lane (two B128 ops, offset applies to
// both the LDS and global side per ISA 10.7/15.18.3). Tracked by ASYNCcnt.
static __device__ __forceinline__ void async_copy_32B(uint32_t lds_addr, const void* gaddr) {
    asm volatile("global_load_async_to_lds_b128 %0, %1, off\n\t"
                 "global_load_async_to_lds_b128 %0, %1, off offset:16"
                 :: "v"(lds_addr), "v"(gaddr) : "memory");
}
static __device__ __forceinline__ void wait_async0() {
    asm volatile("s_wait_asynccnt 0" ::: "memory");
}
// Low 32 bits of a generic pointer to LDS == wave-relative LDS byte address.
static __device__ __forceinline__ uint32_t lds_addr_of(const void* p) {
    return (uint32_t)(uintptr_t)p;
}

// ---------------------------------------------------------------------------
// Kernel 1: O[2048,512] = X[2048,768] @ W[512,768]^T + bias   (z=0 anchor, z=1 pos)
// Block: 256 thr = 8 waves, computes a 64(M) x 64(N) tile; wave -> 16x32 subtile.
// K tiles staged via GLOBAL_LOAD_ASYNC_TO_LDS_B128.
// ---------------------------------------------------------------------------
__global__ __launch_bounds__(256) void k_gemm_bias(
    const float* __restrict__ Xa, const float* __restrict__ Wha, const float* __restrict__ bba,
    float* __restrict__ Oa,
    const float* __restrict__ Xc, const float* __restrict__ Whc, const float* __restrict__ bbc,
    float* __restrict__ Oc)
{
    const float* X; const float* W; const float* bias; float* O;
    if (blockIdx.z == 0) { X = Xa; W = Wha; bias = bba; O = Oa; }
    else                 { X = Xc; W = Whc; bias = bbc; O = Oc; }

    __shared__ float lx[64][33];   // X tile [m][k], padded (8-float chunks stay contiguous)
    __shared__ float lw[64][33];   // W tile [n][k], padded

    const int t    = threadIdx.x;
    const int wave = t >> 5;
    const int lane = t & 31;
    const int half = lane >> 4;    // wave32: lanes 0-15 vs 16-31
    const int lr   = lane & 15;
    const int m0   = (wave >> 1) * 16;
    const int n0   = (wave & 1) * 32;
    const int gm0  = blockIdx.y * 64;
    const int gn0  = blockIdx.x * 64;

    v8f c0 = {}; v8f c1 = {};

    const int lin = t * 8;         // 2048 floats per tile, 8 per thread
    const int lm  = lin >> 5;
    const int lk  = lin & 31;

    const uint32_t ldsx = lds_addr_of(&lx[lm][lk]);
    const uint32_t ldsw = lds_addr_of(&lw[lm][lk]);
    const float* gx = X + (size_t)(gm0 + lm) * EDIM + lk;
    const float* gw = W + (size_t)(gn0 + lm) * EDIM + lk;

    for (int k0 = 0; k0 < EDIM; k0 += 32) {
        __syncthreads();                       // previous tile fully consumed
        async_copy_32B(ldsx, gx + k0);
        async_copy_32B(ldsw, gw + k0);
        wait_async0();
        __syncthreads();                       // all threads' async copies visible
#pragma unroll
        for (int kk = 0; kk < 32; kk += 4) {
            const int ka = kk + half * 2;      // ISA A/B lane layout: half-wave K split
            v2f a, b0, b1;
            a.x  = lx[m0 + lr][ka];      a.y  = lx[m0 + lr][ka + 1];
            b0.x = lw[n0 + lr][ka];      b0.y = lw[n0 + lr][ka + 1];
            b1.x = lw[n0 + 16 + lr][ka]; b1.y = lw[n0 + 16 + lr][ka + 1];
            c0 = wmma_f32(a, b0, c0);
            c1 = wmma_f32(a, b1, c1);
        }
    }

    const float bn0 = bias[gn0 + n0 + lr];
    const float bn1 = bias[gn0 + n0 + 16 + lr];
#pragma unroll
    for (int r = 0; r < 8; ++r) {
        const int m = gm0 + m0 + r + 8 * half; // C/D layout: vgpr r -> rows r / r+8
        O[(size_t)m * HDIM + gn0 + n0 + lr]      = c0[r] + bn0;
        O[(size_t)m * HDIM + gn0 + n0 + 16 + lr] = c1[r] + bn1;
    }
}

// ---------------------------------------------------------------------------
// Kernel 2: per (b,k): logits = (A_b * Watt_k) @ C_b^T  (64x64 via WMMA),
// softmax over 4096 -> att (written to d_out), then out[b,k,h].
// Tiles staged async; Watt scale folded into the A-fragment build.
// ---------------------------------------------------------------------------
__global__ __launch_bounds__(256) void k_attention(
    const float* __restrict__ Aout, const float* __restrict__ Cout,
    const float* __restrict__ Watt, const float* __restrict__ batt,
    float* __restrict__ att_out, float* __restrict__ outvec)
{
    const int k = blockIdx.x;
    const int b = blockIdx.y;

    __shared__ float la[64][33];
    __shared__ float lc[64][33];
    __shared__ float latt[64][65];
    __shared__ float red[256];
    __shared__ float lwk[HDIM];

    const int t    = threadIdx.x;
    const int wave = t >> 5;
    const int lane = t & 31;
    const int half = lane >> 4;
    const int lr   = lane & 15;
    const int m0   = (wave >> 1) * 16;
    const int n0   = (wave & 1) * 32;

    const float* Ab = Aout + (size_t)b * NAA * HDIM;
    const float* Cb = Cout + (size_t)b * NCC * HDIM;
    const float* wk = Watt + (size_t)k * HDIM;

    // stage Watt[k,:] once
    for (int i = t; i < HDIM; i += 256) lwk[i] = wk[i];

    v8f c0 = {}; v8f c1 = {};
    const int lin = t * 8;
    const int li  = lin >> 5;
    const int lh  = lin & 31;

    const uint32_t ldsa = lds_addr_of(&la[li][lh]);
    const uint32_t ldsc = lds_addr_of(&lc[li][lh]);
    const float* ga = Ab + (size_t)li * HDIM + lh;
    const float* gc = Cb + (size_t)li * HDIM + lh;

    for (int h0 = 0; h0 < HDIM; h0 += 32) {
        __syncthreads();
        async_copy_32B(ldsa, ga + h0);
        async_copy_32B(ldsc, gc + h0);
        wait_async0();
        __syncthreads();
#pragma unroll
        for (int kk = 0; kk < 32; kk += 4) {
            const int ka = kk + half * 2;
            const float s0 = lwk[h0 + ka];
            const float s1 = lwk[h0 + ka + 1];
            v2f a, b0, b1;
            a.x  = la[m0 + lr][ka] * s0;  a.y  = la[m0 + lr][ka + 1] * s1;
            b0.x = lc[n0 + lr][ka];       b0.y = lc[n0 + lr][ka + 1];
            b1.x = lc[n0 + 16 + lr][ka];  b1.y = lc[n0 + 16 + lr][ka + 1];
            c0 = wmma_f32(a, b0, c0);
            c1 = wmma_f32(a, b1, c1);
        }
    }
    __syncthreads();
    const float bk = batt[k];
#pragma unroll
    for (int r = 0; r < 8; ++r) {
        latt[m0 + r + 8 * half][n0 + lr]      = c0[r] + bk;
        latt[m0 + r + 8 * half][n0 + 16 + lr] = c1[r] + bk;
    }
    __syncthreads();

    // -------- softmax over all 4096 logits --------
    const int si = t >> 2;           // row
    const int sj = (t & 3) * 16;     // 16 contiguous cols per thread
    float lmax = -3.4e38f;
    for (int e = 0; e < 16; ++e) lmax = fmaxf(lmax, latt[si][sj + e]);
    red[t] = lmax; __syncthreads();
    for (int s = 128; s > 0; s >>= 1) { if (t < s) red[t] = fmaxf(red[t], red[t + s]); __syncthreads(); }
    const float gmax = red[0]; __syncthreads();
    float lsum = 0.f;
    for (int e = 0; e < 16; ++e) {
        float v = __expf(latt[si][sj + e] - gmax);
        latt[si][sj + e] = v; lsum += v;
    }
    red[t] = lsum; __syncthreads();
    for (int s = 128; s > 0; s >>= 1) { if (t < s) red[t] += red[t + s]; __syncthreads(); }
    const float inv = 1.f / red[0]; __syncthreads();

    float* ao = att_out + (size_t)(b * KH + k) * (NAA * NCC);
    for (int e = 0; e < 16; ++e) {
        float v = latt[si][sj + e] * inv;
        latt[si][sj + e] = v;
        ao[si * NCC + sj + e] = v;
    }
    __syncthreads();

    // -------- out[b,k,h] = sum_i A[i,h] * (sum_j att[i,j] * C[j,h]) --------
    const int hh  = t >> 3;          // 0..31 : h within 32-chunk
    const int seg = t & 7;           // i segment of 8
    float* ov = outvec + (size_t)(b * KH + k) * HDIM;
    for (int h0 = 0; h0 < HDIM; h0 += 32) {
        __syncthreads();
        async_copy_32B(ldsa, ga + h0);
        async_copy_32B(ldsc, gc + h0);
        wait_async0();
        __syncthreads();
        float part = 0.f;
        for (int i = seg * 8; i < seg * 8 + 8; ++i) {
            float u = 0.f;
            for (int j = 0; j < 64; ++j) u += latt[i][j] * lc[j][hh];
            part += la[i][hh] * u;
        }
        red[t] = part;
        __syncthreads();
        if (t < 32) {
            float s = 0.f;
#pragma unroll
            for (int q = 0; q < 8; ++q) s += red[t * 8 + q];
            ov[h0 + t] = s;
        }
    }
}

// ---------------------------------------------------------------------------
// Kernel 3: score[b] = outvec[b,:] . Wscore + bscore
// ---------------------------------------------------------------------------
__global__ __launch_bounds__(256) void k_score(
    const float* __restrict__ outvec, const float* __restrict__ Wscore,
    const float* __restrict__ bscore, float* __restrict__ score)
{
    __shared__ float red[256];
    const int b = blockIdx.x, t = threadIdx.x;
    const float* ov = outvec + (size_t)b * (KH * HDIM);
    float s = 0.f;
    for (int i = t; i < KH * HDIM; i += 256) s += ov[i] * Wscore[i];
    red[t] = s; __syncthreads();
    for (int st = 128; st > 0; st >>= 1) { if (t < st) red[t] += red[t + st]; __syncthreads(); }
    if (t == 0) score[b] = red[0] + bscore[0];
}

// ---------------------------------------------------------------------------
// Kernel 4: loss = mean((score - target)^2)
// ---------------------------------------------------------------------------
__global__ void k_loss(const float* __restrict__ score, const float* __restrict__ target,
                       float* __restrict__ loss)
{
    __shared__ float red[32];
    const int t = threadIdx.x;  // 32 threads
    const float d = score[t] - target[t];
    red[t] = d * d;
    __syncthreads();
    for (int s = 16; s > 0; s >>= 1) { if (t < s) red[t] += red[t + s]; __syncthreads(); }
    if (t == 0) loss[0] = red[0] * (1.0f / 32.0f);
}

// ---------------------------------------------------------------------------
extern "C" void kernel_launch(void* const* d_in, const int* in_sizes, int n_in,
                              void* d_out, int out_size, void* d_ws, size_t ws_size,
                              hipStream_t stream)
{
    const float* he_anchor = (const float*)d_in[0];
    const float* he_pos    = (const float*)d_in[1];
    const float* he_neg    = (const float*)d_in[2];
    const float* Wa     = (const float*)d_in[3];
    const float* ba     = (const float*)d_in[4];
    const float* Wc     = (const float*)d_in[5];
    const float* bc     = (const float*)d_in[6];
    const float* Watt   = (const float*)d_in[7];
    const float* batt   = (const float*)d_in[8];
    const float* Wscore = (const float*)d_in[9];
    const float* bscore = (const float*)d_in[10];

    float* out   = (float*)d_out;
    float* score = out;          // [32]
    float* loss  = out + 32;     // [1]
    float* att   = out + 33;     // [32,8,64,64]

    float* ws     = (float*)d_ws;
    float* Aout   = ws;                               // 2048*512
    float* Cout   = Aout + (size_t)2048 * 512;        // 2048*512
    float* outvec = Cout + (size_t)2048 * 512;        // 32*8*512

    k_gemm_bias<<<dim3(8, 32, 2), 256, 0, stream>>>(he_anchor, Wa, ba, Aout,
                                                    he_pos,    Wc, bc, Cout);
    k_attention<<<dim3(8, 32), 256, 0, stream>>>(Aout, Cout, Watt, batt, att, outvec);
    k_score<<<dim3(32), 256, 0, stream>>>(outvec, Wscore, bscore, score);
    k_loss<<<dim3(1), 32, 0, stream>>>(score, he_neg, loss);
}